// SlaterPooling_18451179503905
// MI455X (gfx1250) — compile-verified
//
#include <hip/hip_runtime.h>

// CDNA5 / gfx1250: wave32, WMMA f32 16x16x4 for blocked-LU Schur updates.

typedef __attribute__((ext_vector_type(2))) float v2f;
typedef __attribute__((ext_vector_type(8))) float v8f;

#define NSZ    48          // matrix order (3 * 16)
#define LDA    49          // LDS row stride; gcd(49,64)=1 -> conflict-free
#define NWAVE  4           // matrices (waves) per workgroup
#define NCONF_ 16

__device__ __forceinline__ void wave_lds_fence() {
  // Same-wave cross-lane LDS RAW/WAR ordering: drain DS pipe, fence compiler.
  __builtin_amdgcn_wave_barrier();
  asm volatile("s_wait_dscnt 0" ::: "memory");
  __builtin_amdgcn_wave_barrier();
}

__global__ __launch_bounds__(NWAVE * 32, 1) void
slater_det_kernel(const float* __restrict__ x, const int* __restrict__ idx,
                  float* __restrict__ out, int nmat) {
  __shared__ float sMat[NWAVE][NSZ * LDA];
  __shared__ int   sIdx[NWAVE][NSZ];

  const int lane = threadIdx.x & 31;
  const int wv   = threadIdx.x >> 5;
  const int m    = blockIdx.x * NWAVE + wv;
  if (m >= nmat) return;                 // uniform per wave; no block barriers used
  const int b = m >> 4;                  // NCONF_ == 16
  const int c = m & (NCONF_ - 1);

  float* M = sMat[wv];
  int*   I = sIdx[wv];

  const float* xb = x + (size_t)b * (64 * 32);

  // Prefetch the 8KB x[b] block (32 lanes x 256B strides covers it).
  __builtin_prefetch(xb + lane * 64, 0, 1);

  // Cache the 48 spin-orbital indices of config c.
  I[lane] = idx[c * NSZ + lane];
  if (lane < NSZ - 32) I[lane + 32] = idx[c * NSZ + lane + 32];
  wave_lds_fence();

  // Gather Slater matrix: M[i][j] = x[b, 2*(idx[j]%32) + idx[i]/32, idx[i]%32]
  for (int i = 0; i < NSZ; ++i) {
    const int ii = I[i];
    const float* xrow = xb + (ii & 31);      // orbital column, electron stride 32
    const int eh = ii >> 5;
    for (int j = lane; j < NSZ; j += 32) {
      const int jj = I[j];
      M[i * LDA + j] = xrow[(2 * (jj & 31) + eh) * 32];
    }
  }
  wave_lds_fence();

  double det = 1.0;

  for (int col0 = 0; col0 < NSZ; col0 += 16) {
    const int colEnd = col0 + 16;

    // ---------- panel factorization with partial pivoting ----------
    for (int jc = col0; jc < colEnd; ++jc) {
      // pivot search over rows jc..47 in column jc
      float bestA = -1.0f; int bestR = jc;
      for (int r = jc + lane; r < NSZ; r += 32) {
        float v = fabsf(M[r * LDA + jc]);
        if (v > bestA) { bestA = v; bestR = r; }
      }
      for (int off = 16; off > 0; off >>= 1) {
        float oa  = __shfl_xor(bestA, off, 32);
        int   orr = __shfl_xor(bestR, off, 32);
        if (oa > bestA || (oa == bestA && orr < bestR)) { bestA = oa; bestR = orr; }
      }
      if (bestR != jc) {                     // uniform branch (reduced value)
        for (int j = lane; j < NSZ; j += 32) {
          float t = M[jc * LDA + j];
          M[jc * LDA + j] = M[bestR * LDA + j];
          M[bestR * LDA + j] = t;
        }
        det = -det;
      }
      wave_lds_fence();

      const float piv  = M[jc * LDA + jc];   // broadcast LDS read
      det *= (double)piv;
      const float rpiv = 1.0f / piv;

      // scale sub-column + rank-1 update restricted to the panel
      for (int r = jc + 1 + lane; r < NSZ; r += 32) {
        float l = M[r * LDA + jc] * rpiv;
        M[r * LDA + jc] = l;
        for (int j = jc + 1; j < colEnd; ++j)
          M[r * LDA + j] -= l * M[jc * LDA + j];
      }
      wave_lds_fence();
    }

    const int wTrail = NSZ - colEnd;         // 32, 16, 0
    if (wTrail > 0) {
      // ---------- U12 = L11^{-1} * A12 (unit lower triangular) ----------
      if (lane < wTrail) {
        const int cc = colEnd + lane;        // one trailing column per lane
        float u[16];
        #pragma unroll
        for (int r = 0; r < 16; ++r) u[r] = M[(col0 + r) * LDA + cc];
        #pragma unroll
        for (int r = 1; r < 16; ++r) {
          float acc = u[r];
          for (int t = 0; t < r; ++t)
            acc -= M[(col0 + r) * LDA + (col0 + t)] * u[t];
          u[r] = acc;
        }
        #pragma unroll
        for (int r = 0; r < 16; ++r) M[(col0 + r) * LDA + cc] = u[r];
      }
      wave_lds_fence();                      // re-converged; EXEC all-ones below

      // ---------- A22 -= L21 * U12 via V_WMMA_F32_16X16X4_F32 ----------
      const int n0 = lane & 15;
      const int hi = lane >> 4;              // lane half selects K / row group
      for (int ti = 0; ti < wTrail; ti += 16) {
        for (int tj = 0; tj < wTrail; tj += 16) {
          const int r0 = colEnd + ti;
          const int cb = colEnd + tj;
          v8f cf;
          #pragma unroll
          for (int v = 0; v < 8; ++v)        // C 16x16: VGPR v = row v (+8 hi half)
            cf[v] = M[(r0 + hi * 8 + v) * LDA + cb + n0];
          #pragma unroll
          for (int kk = 0; kk < 16; kk += 4) {
            const int ks = col0 + kk + hi * 2;   // K slice: v0={K0,K2}, v1={K1,K3}
            v2f af, bf;
            af.x = -M[(r0 + n0) * LDA + ks];     // -L21 (f32 WMMA has no A-neg)
            af.y = -M[(r0 + n0) * LDA + ks + 1];
            bf.x =  M[ks * LDA + cb + n0];       // U12 rows striped across lanes
            bf.y =  M[(ks + 1) * LDA + cb + n0];
            cf = __builtin_amdgcn_wmma_f32_16x16x4_f32(
                false, af, false, bf, (short)0, cf, false, false);
          }
          #pragma unroll
          for (int v = 0; v < 8; ++v)
            M[(r0 + hi * 8 + v) * LDA + cb + n0] = cf[v];
        }
      }
      wave_lds_fence();
    }
  }

  if (lane == 0) out[m] = (float)det;
}

extern "C" void kernel_launch(void* const* d_in, const int* in_sizes, int n_in,
                              void* d_out, int out_size, void* d_ws, size_t ws_size,
                              hipStream_t stream) {
  (void)in_sizes; (void)n_in; (void)d_ws; (void)ws_size;
  const float* x   = (const float*)d_in[0];   // (2048, 64, 32) f32
  const int*   idx = (const int*)d_in[1];     // (16, 48) i32
  float* out = (float*)d_out;                 // (2048, 16) f32
  const int nmat = out_size;                  // 32768 determinants
  const int blocks = (nmat + NWAVE - 1) / NWAVE;
  slater_det_kernel<<<blocks, NWAVE * 32, 0, stream>>>(x, idx, out, nmat);
}